// EquivSetConv_4355096839068
// MI455X (gfx1250) — compile-verified
//
#include <hip/hip_runtime.h>

// ---------------------------------------------------------------------------
// EquivSetConv on MI455X (gfx1250), fp32 end-to-end.
//
// Refactored:  concat([X[v], Xe[e]]) @ W2  ==  (X@W2a)[v] + (Xe@W2b)[e]
//              segment_sum(... , vertex)   ==  deg[v]*(X@W2a+b2)[v] + scatter(Xe@W2b rows)
// All dense GEMMs use V_WMMA_F32_16X16X4_F32 (full fp32 precision).
// ---------------------------------------------------------------------------

typedef __attribute__((ext_vector_type(2))) float v2f;
typedef __attribute__((ext_vector_type(8))) float v8f;

#define DFEAT 128

// Y[M x 128] = Xin[M x 128] @ W[128 x 128] (+ bias)
// One wave -> 16 rows x 128 cols (8 WMMA accumulators), 32 k-steps of K=4.
__global__ __launch_bounds__(256) void gemm128_wmma(
    const float* __restrict__ Xin, int Mrows,
    const float* __restrict__ W,     // [128,128] row-major
    const float* __restrict__ bias,  // nullable
    float* __restrict__ Y)
{
    const int lane = threadIdx.x & 31;
    const int wave = threadIdx.x >> 5;
    const int half = lane >> 4;      // 0: lanes 0-15, 1: lanes 16-31
    const int r    = lane & 15;
    const int m0   = (blockIdx.x * 8 + wave) * 16;
    if (m0 >= Mrows) return;         // wave-uniform: EXEC stays all-1s for WMMA

    v8f acc[8] = {};

    // A-fragment row for this lane (clamped so tail waves keep EXEC full)
    int arow = m0 + r;
    if (arow >= Mrows) arow = Mrows - 1;
    const float* ap = Xin + (size_t)arow * DFEAT + 2 * half;

    for (int kk = 0; kk < 32; ++kk) {
        const int k = kk * 4;
        // 16x4 f32 A layout: lane holds K = 2*half, 2*half+1 for M = r
        v2f a;
        a.x = ap[k];
        a.y = ap[k + 1];

        const float* wrow = W + (size_t)(k + 2 * half) * DFEAT + r;
#pragma unroll
        for (int t = 0; t < 8; ++t) {
            // 4x16 f32 B layout: lane holds K = 2*half, 2*half+1 for N = 16*t + r
            v2f b;
            b.x = wrow[t * 16];
            b.y = wrow[t * 16 + DFEAT];
            acc[t] = __builtin_amdgcn_wmma_f32_16x16x4_f32(
                false, a, false, b, (short)0, acc[t], false, false);
        }
    }

    // 16x16 f32 C/D layout: VGPR j holds row m0 + j + 8*half, col 16*t + r
#pragma unroll
    for (int t = 0; t < 8; ++t) {
        const int n = t * 16 + r;
        const float bv = bias ? bias[n] : 0.0f;
#pragma unroll
        for (int j = 0; j < 8; ++j) {
            const int row = m0 + j + 8 * half;
            if (row < Mrows) Y[(size_t)row * DFEAT + n] = acc[t][j] + bv;
        }
    }
}

// dst[sidx[i]] += src[gidx[i]]  (rows of 128 floats); one wave per incidence
// entry, each lane handles a float4 slice. Optionally counts degree.
__global__ __launch_bounds__(256) void scatter_rows(
    const float* __restrict__ src,
    const int* __restrict__ gidx,
    const int* __restrict__ sidx,
    float* __restrict__ dst,
    int* __restrict__ deg,   // nullable
    int nnz)
{
    const int i = blockIdx.x * 8 + (threadIdx.x >> 5);
    if (i >= nnz) return;
    const int lane = threadIdx.x & 31;
    const int g = gidx[i];
    const int s = sidx[i];
    const float4 v = ((const float4*)(src + (size_t)g * DFEAT))[lane];
    float* dp = dst + (size_t)s * DFEAT + lane * 4;
    atomicAdd(dp + 0, v.x);
    atomicAdd(dp + 1, v.y);
    atomicAdd(dp + 2, v.z);
    atomicAdd(dp + 3, v.w);
    if (deg != nullptr && lane == 0) atomicAdd(deg + s, 1);
}

__global__ void zero_kernel(float* __restrict__ p, long n)
{
    long i = (long)blockIdx.x * blockDim.x + threadIdx.x;
    if (i < n) p[i] = 0.0f;
}

// Xn = (1-ALPHA) * (deg[v]*Ap + SB) + ALPHA * X0        (ALPHA = 0.5)
__global__ void blend_kernel(
    const float* __restrict__ Ap, const float* __restrict__ SB,
    const int* __restrict__ deg, const float* __restrict__ X0,
    float* __restrict__ Xn, long n)
{
    long i = (long)blockIdx.x * blockDim.x + threadIdx.x;
    if (i >= n) return;
    const int v = (int)(i >> 7);
    const float d = (float)deg[v];
    Xn[i] = 0.5f * (d * Ap[i] + SB[i]) + 0.5f * X0[i];
}

extern "C" void kernel_launch(void* const* d_in, const int* in_sizes, int n_in,
                              void* d_out, int out_size, void* d_ws, size_t ws_size,
                              hipStream_t stream) {
    const float* X      = (const float*)d_in[0];
    const int*   vertex = (const int*)d_in[1];
    const int*   edges  = (const int*)d_in[2];
    const float* X0     = (const float*)d_in[3];
    const float* W1     = (const float*)d_in[4];
    const float* b1     = (const float*)d_in[5];
    const float* W2     = (const float*)d_in[6];   // [256,128]: W2a = rows 0..127, W2b = rows 128..255
    const float* b2     = (const float*)d_in[7];
    const float* W3     = (const float*)d_in[8];
    const float* b3     = (const float*)d_in[9];
    float* out = (float*)d_out;

    constexpr int N = 50000, M = 25000, NNZ = 800000;
    constexpr size_t ND = (size_t)N * DFEAT;
    constexpr size_t MD = (size_t)M * DFEAT;

    // Workspace layout (all fp32 unless noted), 256B aligned:
    char* ws = (char*)d_ws;
    auto align256 = [](size_t x) { return (x + 255) & ~(size_t)255; };
    size_t off = 0;
    float* XW = (float*)(ws + off); off = align256(off + ND * 4);  // X@W1+b1, reused later for Xn
    float* Xe = (float*)(ws + off); off = align256(off + MD * 4);  // per-edge sums
    float* Ap = (float*)(ws + off); off = align256(off + ND * 4);  // X@W2a + b2
    float* Bm = (float*)(ws + off); off = align256(off + MD * 4);  // Xe@W2b
    float* SB = (float*)(ws + off); off = align256(off + ND * 4);  // scatter of Bm rows to vertices
    int*  deg = (int*)  (ws + off); off = align256(off + (size_t)N * 4);
    float* Xn = XW;  // XW dead after edge scatter -> reuse for blended input

    const int gemmN = (N + 127) / 128;  // 128 rows per 256-thread block
    const int gemmM = (M + 127) / 128;
    const int scatB = (NNZ + 7) / 8;    // 8 incidence entries per block

    // 1) XW = X @ W1 + b1
    gemm128_wmma<<<gemmN, 256, 0, stream>>>(X, N, W1, b1, XW);

    // 2) Xe[e] = sum over incidences of XW[vertex]
    zero_kernel<<<(int)((MD + 255) / 256), 256, 0, stream>>>(Xe, (long)MD);
    scatter_rows<<<scatB, 256, 0, stream>>>(XW, vertex, edges, Xe, nullptr, NNZ);

    // 3) Ap = X @ W2a + b2   (b2 folded here: segment_sum adds it deg[v] times)
    gemm128_wmma<<<gemmN, 256, 0, stream>>>(X, N, W2, b2, Ap);

    // 4) Bm = Xe @ W2b
    gemm128_wmma<<<gemmM, 256, 0, stream>>>(Xe, M, W2 + 128 * DFEAT, nullptr, Bm);

    // 5) SB[v] = sum over incidences of Bm[edges];  deg[v] = incidence count
    zero_kernel<<<(int)((ND + 255) / 256), 256, 0, stream>>>(SB, (long)ND);
    zero_kernel<<<(N + 255) / 256, 256, 0, stream>>>((float*)deg, (long)N);
    scatter_rows<<<scatB, 256, 0, stream>>>(Bm, edges, vertex, SB, deg, NNZ);

    // 6) Xn = 0.5*(deg*Ap + SB) + 0.5*X0
    blend_kernel<<<(int)((ND + 255) / 256), 256, 0, stream>>>(Ap, SB, deg, X0, Xn, (long)ND);

    // 7) out = Xn @ W3 + b3
    gemm128_wmma<<<gemmN, 256, 0, stream>>>(Xn, N, W3, b3, out);
}